// RQAFeatureExtractor_71683004170599
// MI455X (gfx1250) — compile-verified
//
#include <hip/hip_runtime.h>
#include <math.h>

// MI455X / gfx1250, wave32. Strategy:
//  k1: h = relu(x@w1+b1)@w2+b2, store float4{h0,h1,h2,|h|^2}  (512KB, cache-resident)
//  k2: per-batch max d2 via V_WMMA_F32_16X16X4_F32 gram tiles; each wave sweeps 8
//      column tiles against one resident A tile (upper triangle only). Wave/tile
//      indices forced scalar via readfirstlane so guards are s_cbranch, not EXEC masks.
//  k3: per-column serial scan for total/det/vert counts (run-length stat is row-serial)
//  k4: 4->64 head.
// dist is never materialized (would be 2x268MB of HBM traffic); R test uses d2 < thr^2.

typedef __attribute__((ext_vector_type(2))) float v2f;
typedef __attribute__((ext_vector_type(8))) float v8f;

#define BATCH 16
#define SEQ 2048
#define IN_DIM 128
#define ODIM 64
#define EPSF 1e-6f

// ds_swizzle group-of-32 XOR butterflies: offset = xor<<10 | or<<5 | and(0x1f)
template <int PAT>
__device__ __forceinline__ float swz_maxf(float v) {
  return fmaxf(v, __int_as_float(__builtin_amdgcn_ds_swizzle(__float_as_int(v), PAT)));
}
template <int PAT>
__device__ __forceinline__ float swz_addf(float v) {
  return v + __int_as_float(__builtin_amdgcn_ds_swizzle(__float_as_int(v), PAT));
}
template <int PAT>
__device__ __forceinline__ unsigned swz_addu(unsigned v) {
  return v + (unsigned)__builtin_amdgcn_ds_swizzle((int)v, PAT);
}
__device__ __forceinline__ float wave_maxf(float v) {
  v = swz_maxf<0x401F>(v); v = swz_maxf<0x201F>(v); v = swz_maxf<0x101F>(v);
  v = swz_maxf<0x081F>(v); v = swz_maxf<0x041F>(v); return v;
}
__device__ __forceinline__ float wave_addf(float v) {
  v = swz_addf<0x401F>(v); v = swz_addf<0x201F>(v); v = swz_addf<0x101F>(v);
  v = swz_addf<0x081F>(v); v = swz_addf<0x041F>(v); return v;
}
__device__ __forceinline__ unsigned wave_addu(unsigned v) {
  v = swz_addu<0x401F>(v); v = swz_addu<0x201F>(v); v = swz_addu<0x101F>(v);
  v = swz_addu<0x081F>(v); v = swz_addu<0x041F>(v); return v;
}

// ---------------- init: zero accumulators ----------------
__global__ void rqa_init(unsigned* __restrict__ maxd2, unsigned* __restrict__ acc) {
  int i = threadIdx.x;
  if (i < BATCH) maxd2[i] = 0u;
  if (i < BATCH * 3) acc[i] = 0u;
}

// ---------------- k1: embedding, one wave per row ----------------
__global__ void rqa_embed(const float* __restrict__ x,
                          const float* __restrict__ w1, const float* __restrict__ b1,
                          const float* __restrict__ w2, const float* __restrict__ b2,
                          float4* __restrict__ hbuf) {
  const int wave = __builtin_amdgcn_readfirstlane(threadIdx.x >> 5);  // scalar
  const int lane = threadIdx.x & 31;
  const int row  = blockIdx.x * 8 + wave;              // scalar, 0..B*S-1
  // lane l loads x[row, 4l..4l+3]: fully coalesced 512B per wave, SGPR base
  const float4 xv = ((const float4*)(x + (size_t)row * IN_DIM))[lane];
  const float* wr = w1 + (lane * 4) * 6;               // w1[128,6] row-major, 3KB hot
  float p[6];
#pragma unroll
  for (int j = 0; j < 6; ++j)
    p[j] = xv.x * wr[j] + xv.y * wr[6 + j] + xv.z * wr[12 + j] + xv.w * wr[18 + j];
#pragma unroll
  for (int j = 0; j < 6; ++j) p[j] = wave_addf(p[j]);
  if (lane == 0) {
    float g[6];
#pragma unroll
    for (int j = 0; j < 6; ++j) g[j] = fmaxf(p[j] + b1[j], 0.0f);
    float h0 = b2[0], h1 = b2[1], h2 = b2[2];
#pragma unroll
    for (int j = 0; j < 6; ++j) {
      h0 += g[j] * w2[j * 3 + 0];
      h1 += g[j] * w2[j * 3 + 1];
      h2 += g[j] * w2[j * 3 + 2];
    }
    float sq = h0 * h0 + h1 * h1 + h2 * h2;
    hbuf[row] = make_float4(h0, h1, h2, sq);           // .w carries |h|^2
  }
}

// ---------------- k2: max d2 via WMMA gram tiles, 8 tiles per wave ----------------
// A (16x4 f32): lanes 0-15 hold row M=lane K0(v0)/K1(v1); lanes 16-31 hold K2/K3.
// B (4x16 f32): same per-lane gather with columns. K=3 padded with 0.
// C (16x16 f32): c[v]: lanes 0-15 -> D[M=v][N=lane]; lanes 16-31 -> D[M=v+8][N=lane-16].
__global__ void __launch_bounds__(256) rqa_maxd2(const float4* __restrict__ hbuf,
                                                 unsigned* __restrict__ maxd2) {
  __shared__ float smax[8];
  const int wave = __builtin_amdgcn_readfirstlane(threadIdx.x >> 5);  // scalar
  const int lane = threadIdx.x & 31;
  const int b    = blockIdx.y;
  const int idx  = blockIdx.x * 8 + wave;              // scalar, 0..2047 per batch
  const int ti   = idx >> 4;                           // scalar row tile 0..127
  const int g    = idx & 15;                           // scalar column-tile group
  const int  l16 = lane & 15;
  const bool hi  = lane >= 16;
  const float4* hb = hbuf + b * SEQ;
  float m = 0.0f;
  if (g * 8 + 7 >= ti) {                               // scalar branch
    const float4 ra = hb[ti * 16 + l16];
    v2f a; a[0] = hi ? ra.z : ra.x; a[1] = hi ? 0.0f : ra.y;  // K3 pad (.w is |h|^2!)
    const int rowoff = ti * 16 + (hi ? 8 : 0);
    float sqr[8];
#pragma unroll
    for (int v = 0; v < 8; ++v) sqr[v] = hb[rowoff + v].w;
#pragma unroll
    for (int k = 0; k < 8; ++k) {
      const int tj = g * 8 + k;
      if (tj < ti) continue;                           // scalar branch: EXEC untouched
      const float4 rb = hb[tj * 16 + l16];
      v2f bm; bm[0] = hi ? rb.z : rb.x; bm[1] = hi ? 0.0f : rb.y;
      v8f c = {};
      c = __builtin_amdgcn_wmma_f32_16x16x4_f32(false, a, false, bm, (short)0, c,
                                                false, false);
      const float sqc = rb.w;
#pragma unroll
      for (int v = 0; v < 8; ++v) m = fmaxf(m, sqr[v] + sqc - 2.0f * c[v]);
    }
  }
  m = wave_maxf(m);
  if (lane == 0) smax[wave] = m;
  __syncthreads();
  if (threadIdx.x == 0) {
    float bmax = smax[0];
#pragma unroll
    for (int w = 1; w < 8; ++w) bmax = fmaxf(bmax, smax[w]);
    // d2 >= 0 after clamp, so uint bit-pattern atomicMax is order-preserving
    atomicMax(maxd2 + b, __float_as_uint(fmaxf(bmax, 0.0f)));
  }
}

// ---------------- k3: per-column scan for total / det / vert ----------------
__global__ void __launch_bounds__(256) rqa_stats(const float4* __restrict__ hbuf,
                                                 const unsigned* __restrict__ maxd2,
                                                 const float* __restrict__ theta,
                                                 unsigned* __restrict__ acc) {
  __shared__ unsigned sred[8][3];
  const int b = blockIdx.x >> 3;                        // 8 blocks of 256 cols per batch
  const int t = ((blockIdx.x & 7) << 8) + threadIdx.x;  // column
  const int chunk = blockIdx.y;                         // 4 row-chunks of 512
  const float4* hb = hbuf + b * SEQ;
  const float4 ht = hb[t];
  const float md2 = __uint_as_float(maxd2[b]);
  const float sig = 1.0f / (1.0f + expf(-theta[0]));
  const float thr = sig * sqrtf(md2);
  const float thr2 = thr * thr;
  const int r0 = chunk * 512, r1 = r0 + 512;
  const int istart = (r0 == 0) ? 0 : r0 - 1;            // halo row for run detection
  const int iend   = (r1 >= SEQ) ? SEQ : r1 + 1;
  unsigned total = 0, det = 0, vert = 0;
  unsigned rp = 0, rpp = 0;                             // R[i-1], R[i-2]
#pragma unroll 4
  for (int i = istart; i < iend; ++i) {
    const float4 hr = hb[i];                            // wave-uniform load -> broadcast
    float d2 = hr.w + ht.w - 2.0f * (ht.x * hr.x + ht.y * hr.y + ht.z * hr.z);
    d2 = fmaxf(d2, 0.0f);                               // matches sqrt(max(d2,0)) semantics
    const unsigned r = (d2 < thr2) ? 1u : 0u;
    if (i >= r0 && i < r1) {                            // uniform branch
      total += r;
      const int off = t - i;                            // superdiagonals 1..9
      det += (r & ((off >= 1 && off <= 9) ? 1u : 0u));
    }
    const int im1 = i - 1;                              // run of len>=2 starts at i-1
    if (im1 >= r0 && im1 < r1) vert += rp * (1u - rpp) * r;
    rpp = rp; rp = r;
  }
  total = wave_addu(total);
  det   = wave_addu(det);
  vert  = wave_addu(vert);
  const int wave = threadIdx.x >> 5;
  if ((threadIdx.x & 31) == 0) {
    sred[wave][0] = total; sred[wave][1] = det; sred[wave][2] = vert;
  }
  __syncthreads();
  if (threadIdx.x == 0) {
    unsigned s0 = 0, s1 = 0, s2 = 0;
#pragma unroll
    for (int w = 0; w < 8; ++w) { s0 += sred[w][0]; s1 += sred[w][1]; s2 += sred[w][2]; }
    atomicAdd(acc + b * 3 + 0, s0);
    atomicAdd(acc + b * 3 + 1, s1);
    atomicAdd(acc + b * 3 + 2, s2);
  }
}

// ---------------- k4: metrics -> 4x64 head ----------------
__global__ void rqa_head(const unsigned* __restrict__ acc,
                         const float* __restrict__ w3, const float* __restrict__ b3,
                         float* __restrict__ out) {
  const int b = blockIdx.x, o = threadIdx.x;
  const float total = (float)acc[b * 3 + 0];
  const float diag  = (float)acc[b * 3 + 1];
  const float vert  = (float)acc[b * 3 + 2];
  const float rr   = total * (1.0f / ((float)SEQ * (float)SEQ));
  const float det  = diag / (total + EPSF);
  const float lam  = vert / (total + EPSF);
  const float entr = -total * logf(1.0f + EPSF);        // R is 0/1 -> closed form
  float v = rr * w3[0 * ODIM + o] + det * w3[1 * ODIM + o] +
            lam * w3[2 * ODIM + o] + entr * w3[3 * ODIM + o] + b3[o];
  out[b * ODIM + o] = fmaxf(v, 0.0f);
}

extern "C" void kernel_launch(void* const* d_in, const int* in_sizes, int n_in,
                              void* d_out, int out_size, void* d_ws, size_t ws_size,
                              hipStream_t stream) {
  const float* x     = (const float*)d_in[0];
  const float* theta = (const float*)d_in[1];
  const float* w1    = (const float*)d_in[2];
  const float* b1    = (const float*)d_in[3];
  const float* w2    = (const float*)d_in[4];
  const float* b2    = (const float*)d_in[5];
  const float* w3    = (const float*)d_in[6];
  const float* b3    = (const float*)d_in[7];
  float* out = (float*)d_out;

  char* ws = (char*)d_ws;
  float4*   hbuf  = (float4*)ws;                        // 16*2048*16 = 524288 B
  unsigned* maxd2 = (unsigned*)(ws + 524288);           // 16 u32
  unsigned* acc   = (unsigned*)(ws + 524288 + 256);     // 16*3 u32

  rqa_init <<<1, 64, 0, stream>>>(maxd2, acc);
  rqa_embed<<<(BATCH * SEQ) / 8, 256, 0, stream>>>(x, w1, b1, w2, b2, hbuf);
  rqa_maxd2<<<dim3(2048 / 8, BATCH), 256, 0, stream>>>(hbuf, maxd2);
  rqa_stats<<<dim3((BATCH * SEQ) / 256, 4), 256, 0, stream>>>(hbuf, maxd2, theta, acc);
  rqa_head <<<BATCH, ODIM, 0, stream>>>(acc, w3, b3, out);
}